// Spatial_15968688406784
// MI455X (gfx1250) — compile-verified
//
#include <hip/hip_runtime.h>
#include <stdint.h>

// Problem constants (fixed by the reference).
#define N_    64
#define C_    64
#define HIN   110   // unpadded height/width of x and out
#define HPAD  112   // mask height/width (x zero-padded to this)
#define PS    4     // pooling window
#define NBH   28    // HPAD / PS block-rows
#define BHALF 14    // block-rows per workgroup (2 workgroups per plane)

// ---- CDNA5 async global->LDS copy (gfx1250), tracked by ASYNCcnt ----
__device__ __forceinline__ void async_ld_b128(void* lds_dst, const void* gsrc) {
    unsigned lptr = (unsigned)(uintptr_t)lds_dst;               // low 32 bits = LDS offset
    unsigned long long gptr = (unsigned long long)(uintptr_t)gsrc;
    asm volatile("global_load_async_to_lds_b128 %0, %1, off"
                 :: "v"(lptr), "v"(gptr)
                 : "memory");
}

__device__ __forceinline__ void wait_async0() {
    asm volatile("s_wait_asynccnt 0" ::: "memory");
}

__global__ void __launch_bounds__(128)
spatial_pool_mask_kernel(const float* __restrict__ x,
                         const float* __restrict__ mask,
                         float* __restrict__ out)
{
    const int plane = blockIdx.y;                 // n*C + c
    const int bh0   = blockIdx.x * BHALF;         // first block-row for this WG
    const int bh1   = bh0 + BHALF;

    const float* xp = x    + (size_t)plane * (HIN * HIN);
    const float* mp = mask + (size_t)plane * (HPAD * HPAD);
    float*       op = out  + (size_t)plane * (HIN * HIN);

    // Double-buffered strips: x strip = up to 4*110 floats, mask strip = 4*112.
    __shared__ alignas(16) float sx[2][448];
    __shared__ alignas(16) float sm[2][448];

    const int tid = threadIdx.x;                  // 0..127
    const int col = tid;                          // one lane per padded column

    auto issue = [&](int bh, int buf) {
        const int r0  = bh * PS;
        // number of float4s in the contiguous x strip (440 floats, or 220 at the edge)
        const int nx4 = (r0 + PS <= HIN) ? (PS * HIN / 4) : ((HIN - r0) * HIN / 4);
        if (tid < nx4)
            async_ld_b128(&sx[buf][tid * 4], xp + (size_t)r0 * HIN + tid * 4);
        if (tid < HPAD)
            async_ld_b128(&sm[buf][tid * 4], mp + (size_t)r0 * HPAD + tid * 4);
    };

    issue(bh0, 0);
    for (int bh = bh0; bh < bh1; ++bh) {
        const int buf = (bh - bh0) & 1;

        wait_async0();            // this wave's async writes for strip bh are in LDS
        __syncthreads();          // ...and every other wave's too; also orders all
                                  // waves' compute(bh-1) LDS reads before the re-fill
                                  // of buf^1 issued below (single barrier suffices:
                                  // DS reads are consumed before each wave arrives)

        if (bh + 1 < bh1) issue(bh + 1, buf ^ 1);   // DMA next strip || compute this one

        const int r0 = bh * PS;
        float xv[PS];
        float s = 0.0f;
        #pragma unroll
        for (int rr = 0; rr < PS; ++rr) {
            const bool valid = (r0 + rr < HIN) && (col < HIN);   // padded region of x is 0
            float v = 0.0f;
            if (valid) {
                v = sx[buf][rr * HIN + col];
                s += v * sm[buf][rr * HPAD + col];
            }
            xv[rr] = v;
        }
        // 4x4 window sum: lanes of a window are 4-aligned inside a wave32.
        s += __shfl_xor(s, 1, 32);
        s += __shfl_xor(s, 2, 32);
        const float b = (s > 0.0f) ? 1.0f : 0.0f;

        #pragma unroll
        for (int rr = 0; rr < PS; ++rr) {
            if ((r0 + rr < HIN) && (col < HIN))
                op[(size_t)(r0 + rr) * HIN + col] = xv[rr] * b;   // coalesced b32 stores
        }
    }
}

extern "C" void kernel_launch(void* const* d_in, const int* in_sizes, int n_in,
                              void* d_out, int out_size, void* d_ws, size_t ws_size,
                              hipStream_t stream) {
    (void)in_sizes; (void)n_in; (void)out_size; (void)d_ws; (void)ws_size;
    const float* x    = (const float*)d_in[0];
    const float* mask = (const float*)d_in[1];
    // d_in[2] is p_size == 4, baked in as a compile-time constant.
    float* out = (float*)d_out;

    dim3 grid(NBH / BHALF, N_ * C_);   // (2, 4096) workgroups
    spatial_pool_mask_kernel<<<grid, 128, 0, stream>>>(x, mask, out);
}